// GATLayer_27960237097247
// MI455X (gfx1250) — compile-verified
//
#include <hip/hip_runtime.h>
#include <hip/hip_bf16.h>
#include <math.h>

typedef __attribute__((ext_vector_type(2))) float v2f;
typedef __attribute__((ext_vector_type(8))) float v8f;

#define B_DIM 2
#define N_DIM 1024
#define C_IN  128
#define H_DIM 16
#define CPH   8
#define C_OUT 128
#define ALPHA 0.2f
#define NEG_BIG (-9.0e15f)

// GEMM block tile: 64x128, K-chunk 32; 8 waves, each owning a 32x32 (2x2 WMMA) tile.
#define TM 64
#define TN 128
#define TK 32
#define A_PAD 36   // floats per A row  (144B, 16B-aligned, conflict-free frag reads)
#define B_PAD 132  // floats per B row  (528B, 16B-aligned)

// ---------------------------------------------------------------------------
// Tiled fp32 GEMM with bias + optional sigmoid on V_WMMA_F32_16X16X4_F32.
// Double-buffered: K-tiles staged to LDS via GLOBAL_LOAD_ASYNC_TO_LDS_B128;
// the next tile's 6 per-wave async DMAs are in flight while the current
// tile's 32 WMMAs execute (s_wait_asynccnt 0x6 = keep newest 6 outstanding;
// async loads complete in order, so the oldest 6 == current buffer are done).
// A-frag (16x4 f32): lanes 0-15 hold K=0(v0),K=1(v1); lanes 16-31 K=2,K=3.
// B-frag (4x16 f32): same K split, N striped across lanes.
// C/D (16x16 f32):   VGPR r -> row r (lanes 0-15) / row r+8 (lanes 16-31).
// ---------------------------------------------------------------------------
__global__ __launch_bounds__(256)
void gemm_bias_act(const float* __restrict__ A, const float* __restrict__ Bm,
                   const float* __restrict__ bias, float* __restrict__ C,
                   int M, int N, int K, int act)
{
    __shared__ float As[2][TM][A_PAD];
    __shared__ float Bs[2][TK][B_PAD];

    const int t     = threadIdx.x;
    const int lane  = t & 31;
    const int wave  = t >> 5;       // 0..7
    const int waveM = wave >> 2;    // 0..1  (32-row slabs)
    const int waveN = wave & 3;     // 0..3  (32-col slabs)
    const int row0  = blockIdx.y * TM;
    const int col0  = blockIdx.x * TN;

    // Issue the 6 per-wave async global->LDS DMAs for one K-tile into buffer p.
    auto issue_tile = [&](int p, int k0) {
        // A tile: 64 rows x 32 floats = 512 x 16B chunks (2 per thread)
        for (int c = t; c < (TM * TK) / 4; c += 256) {
            int m  = c >> 3;           // TK/4 = 8 chunks per row
            int kq = c & 7;
            unsigned lds = (unsigned)(size_t)&As[p][m][kq * 4];
            unsigned gof = (unsigned)(((size_t)(row0 + m) * K + k0 + kq * 4) * 4u);
            asm volatile("global_load_async_to_lds_b128 %0, %1, %2"
                         :: "v"(lds), "v"(gof), "s"(A) : "memory");
        }
        // B tile: 32 rows x 128 floats = 1024 x 16B chunks (4 per thread)
        for (int c = t; c < (TK * TN) / 4; c += 256) {
            int k  = c >> 5;           // TN/4 = 32 chunks per row
            int nq = c & 31;
            unsigned lds = (unsigned)(size_t)&Bs[p][k][nq * 4];
            unsigned gof = (unsigned)(((size_t)(k0 + k) * N + col0 + nq * 4) * 4u);
            asm volatile("global_load_async_to_lds_b128 %0, %1, %2"
                         :: "v"(lds), "v"(gof), "s"(Bm) : "memory");
        }
    };

    v8f acc00 = {}, acc01 = {}, acc10 = {}, acc11 = {};

    const int l15   = lane & 15;
    const int mrow0 = waveM * 32 + l15;
    const int mrow1 = mrow0 + 16;
    const int ncol0 = waveN * 32 + l15;
    const int ncol1 = ncol0 + 16;
    const int ksel  = (lane < 16) ? 0 : 2;

    issue_tile(0, 0);                         // prologue
    int p = 0;
    for (int k0 = 0; k0 < K; k0 += TK) {
        const bool hasNext = (k0 + TK) < K;
        if (hasNext) {
            issue_tile(p ^ 1, k0 + TK);       // overlap next tile's DMA w/ compute
            asm volatile("s_wait_asynccnt 0x6" ::: "memory");
        } else {
            asm volatile("s_wait_asynccnt 0x0" ::: "memory");
        }
        __syncthreads();

        const float (*Asp)[A_PAD] = As[p];
        const float (*Bsp)[B_PAD] = Bs[p];
#pragma unroll
        for (int kk = 0; kk < TK; kk += 4) {
            const int kb = kk + ksel;
            v2f a0, a1, b0, b1;
            a0[0] = Asp[mrow0][kb];    a0[1] = Asp[mrow0][kb + 1];
            a1[0] = Asp[mrow1][kb];    a1[1] = Asp[mrow1][kb + 1];
            b0[0] = Bsp[kb][ncol0];    b0[1] = Bsp[kb + 1][ncol0];
            b1[0] = Bsp[kb][ncol1];    b1[1] = Bsp[kb + 1][ncol1];
            acc00 = __builtin_amdgcn_wmma_f32_16x16x4_f32(false, a0, false, b0, (short)0, acc00, false, false);
            acc01 = __builtin_amdgcn_wmma_f32_16x16x4_f32(false, a0, false, b1, (short)0, acc01, false, false);
            acc10 = __builtin_amdgcn_wmma_f32_16x16x4_f32(false, a1, false, b0, (short)0, acc10, false, false);
            acc11 = __builtin_amdgcn_wmma_f32_16x16x4_f32(false, a1, false, b1, (short)0, acc11, false, false);
        }
        __syncthreads();                      // frees buffer p for reuse
        p ^= 1;
    }

    const int moff = (lane < 16) ? 0 : 8;
    const int nA = col0 + waveN * 32 + l15;        // ncol0 global
    const int nB = nA + 16;                        // ncol1 global
    const float bvA = bias[nA];
    const float bvB = bias[nB];
#pragma unroll
    for (int sub = 0; sub < 4; ++sub) {
        const v8f* accp = (sub == 0) ? &acc00 : (sub == 1) ? &acc01 : (sub == 2) ? &acc10 : &acc11;
        const int n  = (sub & 1) ? nB : nA;
        const float bv = (sub & 1) ? bvB : bvA;
        const int mb = row0 + waveM * 32 + ((sub >> 1) * 16) + moff;
#pragma unroll
        for (int r = 0; r < 8; ++r) {
            float v = (*accp)[r] + bv;
            if (act == 1)  // sigmoid via fast rcp (threshold at 0.5 tolerates 1ulp)
                v = __builtin_amdgcn_rcpf(1.0f + __expf(-v));
            C[(size_t)(mb + r) * N + n] = v;
        }
    }
}

// ---------------------------------------------------------------------------
// lp[row,h] = sum_c x[row,h,c]*a[h,c];  lc[row,h] = sum_c x[row,h,c]*a[h,8+c]
// ---------------------------------------------------------------------------
__global__ __launch_bounds__(256)
void logits_kernel(const float* __restrict__ x, const float* __restrict__ a,
                   float* __restrict__ lp, float* __restrict__ lc)
{
    int idx = blockIdx.x * 256 + threadIdx.x;       // row*16 + h
    if (idx >= B_DIM * N_DIM * H_DIM) return;
    int row = idx >> 4, h = idx & 15;
    const float* xr = x + (size_t)row * C_OUT + h * CPH;
    float p = 0.0f, c = 0.0f;
#pragma unroll
    for (int cc = 0; cc < CPH; ++cc) {
        p += xr[cc] * a[h * (2 * CPH) + cc];
        c += xr[cc] * a[h * (2 * CPH) + CPH + cc];
    }
    lp[idx] = p;
    lc[idx] = c;
}

// ---------------------------------------------------------------------------
// adjbits[b,i,j/32] bit j%32 = ((e[b,i,j]+e[b,j,i]) > 1.0)   (== (e+eT)/2 > .5)
// ---------------------------------------------------------------------------
__global__ __launch_bounds__(256)
void adj_mask_kernel(const float* __restrict__ e, unsigned* __restrict__ adjbits)
{
    __shared__ float tileT[32][33];
    __shared__ unsigned bits[32];

    const int b  = blockIdx.z;
    const int i0 = blockIdx.y * 32;
    const int j0 = blockIdx.x * 32;
    const int t  = threadIdx.x;
    const float* eb = e + (size_t)b * N_DIM * N_DIM;

    for (int idx = t; idx < 1024; idx += 256) {
        int jj = idx >> 5, ii = idx & 31;              // coalesced over ii
        tileT[ii][jj] = eb[(size_t)(j0 + jj) * N_DIM + (i0 + ii)];
    }
    if (t < 32) bits[t] = 0u;
    __syncthreads();

    for (int idx = t; idx < 1024; idx += 256) {
        int r = idx >> 5, jj = idx & 31;               // coalesced over jj
        float v = eb[(size_t)(i0 + r) * N_DIM + (j0 + jj)] + tileT[r][jj];
        if (v > 1.0f) atomicOr(&bits[r], 1u << jj);
    }
    __syncthreads();
    if (t < 32)
        adjbits[((size_t)b * N_DIM + i0 + t) * (N_DIM / 32) + (j0 >> 5)] = bits[t];
}

// ---------------------------------------------------------------------------
// Fused masked-softmax + aggregation; one 128-thread workgroup per (b,i).
// ---------------------------------------------------------------------------
__global__ __launch_bounds__(128)
void attn_kernel(const float* __restrict__ x, const float* __restrict__ lp,
                 const float* __restrict__ lc, const unsigned* __restrict__ adjbits,
                 float* __restrict__ out)
{
    __shared__ unsigned maskRow[32];
    __shared__ float lps[16];
    __shared__ float redM[8][16];
    __shared__ float redS[8][16];
    __shared__ float mFin[16];
    __shared__ float sInv[16];

    const int bi = blockIdx.x;          // b*N + i
    const int b  = bi >> 10;
    const int t  = threadIdx.x;

    if (t < 32) maskRow[t] = adjbits[(size_t)bi * 32 + t];
    if (t < 16) lps[t] = lp[(size_t)bi * 16 + t];
    __syncthreads();

    {   // ---- pass A: per-head max & sumexp ----
        const int h = t & 15, jg = t >> 4;
        const float lpv = lps[h];
        const float* lcb = lc + (size_t)b * N_DIM * H_DIM;
        float m = -INFINITY, s = 0.0f;
        for (int k = 0; k < N_DIM / 8; ++k) {
            int j = jg * (N_DIM / 8) + k;
            float l;
            if ((maskRow[j >> 5] >> (j & 31)) & 1u) {
                float lg = lpv + lcb[j * H_DIM + h];
                l = (lg > 0.0f) ? lg : ALPHA * lg;
            } else {
                l = NEG_BIG;
            }
            float mn = fmaxf(m, l);
            s = s * __expf(m - mn) + __expf(l - mn);
            m = mn;
        }
        redM[jg][h] = m;
        redS[jg][h] = s;
    }
    __syncthreads();
    if (t < 16) {
        float M = redM[0][t], S = redS[0][t];
#pragma unroll
        for (int g = 1; g < 8; ++g) {
            float m2 = redM[g][t], s2 = redS[g][t];
            float mn = fmaxf(M, m2);
            S = S * __expf(M - mn) + s2 * __expf(m2 - mn);
            M = mn;
        }
        mFin[t] = M;
        sInv[t] = 1.0f / S;
    }
    __syncthreads();

    {   // ---- pass B: weighted aggregation ----
        const int h = t >> 3;
        const float lpv = lps[h], M = mFin[h], is = sInv[h];
        const float* lcb = lc + (size_t)b * N_DIM * H_DIM;
        const float* xb  = x  + (size_t)b * N_DIM * C_OUT;
        float acc = 0.0f;
        for (int j = 0; j < N_DIM; ++j) {
            float l;
            if ((maskRow[j >> 5] >> (j & 31)) & 1u) {
                float lg = lpv + lcb[j * H_DIM + h];
                l = (lg > 0.0f) ? lg : ALPHA * lg;
            } else {
                l = NEG_BIG;
            }
            float p = __expf(l - M) * is;
            acc += p * xb[(size_t)j * C_OUT + t];    // coalesced across t
        }
        out[(size_t)bi * C_OUT + t] = acc;
    }
}

// ---------------------------------------------------------------------------
extern "C" void kernel_launch(void* const* d_in, const int* in_sizes, int n_in,
                              void* d_out, int out_size, void* d_ws, size_t ws_size,
                              hipStream_t stream)
{
    const float* node_feats = (const float*)d_in[0];   // [2,1024,128]
    const float* edges      = (const float*)d_in[1];   // [2,1024,1024]
    const float* W_proj     = (const float*)d_in[2];   // [128,128]
    const float* b_proj     = (const float*)d_in[3];   // [128]
    const float* a          = (const float*)d_in[4];   // [16,16]
    const float* W_adj      = (const float*)d_in[5];   // [1024,1024]
    const float* b_adj      = (const float*)d_in[6];   // [1024]
    float* out = (float*)d_out;                        // [2,1024,128]

    // workspace layout (floats)
    float* e  = (float*)d_ws;                          // 2048*1024      (8 MB)
    float* x  = e  + (size_t)2048 * 1024;              // 2048*128       (1 MB)
    float* lp = x  + (size_t)2048 * 128;               // 2048*16
    float* lc = lp + (size_t)2048 * 16;                // 2048*16
    unsigned* adjbits = (unsigned*)(lc + (size_t)2048 * 16);   // 2048*32 words

    const int M = B_DIM * N_DIM;                       // 2048

    // x = node_feats @ W_proj + b_proj            (WMMA fp32)
    gemm_bias_act<<<dim3(C_OUT / TN, M / TM), 256, 0, stream>>>(
        node_feats, W_proj, b_proj, x, M, C_OUT, C_IN, /*act=*/0);

    // e = sigmoid(edges @ W_adj + b_adj)          (WMMA fp32, dominant GEMM)
    gemm_bias_act<<<dim3(N_DIM / TN, M / TM), 256, 0, stream>>>(
        edges, W_adj, b_adj, e, M, N_DIM, N_DIM, /*act=*/1);

    // lp/lc per (row, head)
    logits_kernel<<<(M * H_DIM + 255) / 256, 256, 0, stream>>>(x, a, lp, lc);

    // adjacency bitmask from symmetrized, thresholded e
    adj_mask_kernel<<<dim3(N_DIM / 32, N_DIM / 32, B_DIM), 256, 0, stream>>>(e, adjbits);

    // fused masked softmax + aggregation
    attn_kernel<<<M, 128, 0, stream>>>(x, lp, lc, adjbits, out);
}